// CopyNet_71416716198400
// MI455X (gfx1250) — compile-verified
//
#include <hip/hip_runtime.h>
#include <hip/hip_bf16.h>

#ifndef __has_builtin
#define __has_builtin(x) 0
#endif

#define HH 768
#define BB 2
#define SS 512
#define TT_ 128

// fused kernel tile sizes
#define TS 32          // src rows per block
#define TTI 16         // tgt rows per block
#define RPITCH 780     // 768 + 3*4 pad floats (TDM pad: 4 DWORDs per 256 DWORDs)
#define SEGP 260       // 256 + 4 pad floats per segment

typedef __attribute__((ext_vector_type(2))) float v2f;
typedef __attribute__((ext_vector_type(8))) float v8f;
typedef unsigned int u32;
typedef __attribute__((ext_vector_type(4))) u32 u32x4;
typedef __attribute__((ext_vector_type(8))) int i32x8;
typedef __attribute__((ext_vector_type(4))) int i32x4;

#if __has_builtin(__builtin_amdgcn_tensor_load_to_lds) && __has_builtin(__builtin_amdgcn_s_wait_tensorcnt)
#define USE_TDM 1
#else
#define USE_TDM 0
#endif

__device__ __forceinline__ float htanh(float x) {
#if __has_builtin(__builtin_amdgcn_tanhf)
  return __builtin_amdgcn_tanhf(x);
#else
  return tanhf(x);
#endif
}

// ---------------------------------------------------------------------------
// Kernel 1: Y[M,D] = X[M,H] * W[D,H]^T  via V_WMMA_F32_16X16X4_F32.
// One wave computes one 16x16 tile; K-loop over H in steps of 4.
// A 16x4 layout: lanes 0-15 = rows M0..M15 holding K={k,k+1} in VGPR0/1,
// lanes 16-31 = same rows holding K={k+2,k+3}. B mirrored (cols across lanes).
// C/D: VGPR j -> row (j + 8*(lane>=16)), col = lane&15.
// ---------------------------------------------------------------------------
__global__ __launch_bounds__(32) void linear_wmma_f32(
    const float* __restrict__ X, const float* __restrict__ W,
    float* __restrict__ Y, int M, int D) {
  const int lane = threadIdx.x;
  const int half = lane >> 4;
  const int l16  = lane & 15;
  const int m0 = blockIdx.x * 16;
  const int n0 = blockIdx.y * 16;
  const float* xrow = X + (size_t)(m0 + l16) * HH + 2 * half;
  const float* wrow = W + (size_t)(n0 + l16) * HH + 2 * half;
  v8f c = {0.f, 0.f, 0.f, 0.f, 0.f, 0.f, 0.f, 0.f};
#pragma unroll 8
  for (int k = 0; k < HH; k += 4) {
    v2f a = *(const v2f*)(xrow + k);
    v2f b = *(const v2f*)(wrow + k);
    c = __builtin_amdgcn_wmma_f32_16x16x4_f32(
        /*neg_a=*/false, a, /*neg_b=*/false, b,
        /*c_mod=*/(short)0, c, /*reuse_a=*/false, /*reuse_b=*/false);
  }
  float* y = Y + (size_t)(m0 + 8 * half) * D + n0 + l16;
#pragma unroll
  for (int j = 0; j < 8; ++j) y[(size_t)j * D] = c[j];
}

#if USE_TDM
// 2D tile load via Tensor Data Mover: rows_tile rows of 768 f32, row stride
// 768 f32 in memory, padded in LDS with 4 DWORDs after every 256 DWORDs
// (pad_interval=7 -> 256 DW, pad_amount=3 -> 4 DW) => LDS row pitch 780 f32.
__device__ __forceinline__ void tdm_load_2d(const void* gptr, u32 lds_byte_addr,
                                            u32 rows_tile, u32 rows_tensor) {
  unsigned long long ga = (unsigned long long)gptr;
  u32x4 g0;
  g0.x = 0x1u;                              // count=1, user descriptor
  g0.y = lds_byte_addr;                     // lds_addr
  g0.z = (u32)ga;                           // global_addr[31:0]
  g0.w = (u32)(ga >> 32) | (2u << 30);      // global_addr[56:32] | type=2
  i32x8 g1;
  g1[0] = (int)((2u << 16)      // data_size = 4B
              | (1u << 20)      // pad_enable
              | (7u << 22)      // pad_interval: 256 DWORDs
              | (3u << 25));    // pad_amount: 4 DWORDs
  g1[1] = (int)((u32)HH << 16);             // tensor_dim0[15:0]
  g1[2] = (int)((rows_tensor & 0xFFFFu) << 16); // tensor_dim1[15:0]
  g1[3] = (int)((u32)HH << 16);             // tile_dim0 = 768
  g1[4] = (int)(rows_tile & 0xFFFFu);       // tile_dim1
  g1[5] = HH;                               // tensor_dim0_stride = 768
  g1[6] = 0;
  g1[7] = 0;
  i32x4 z4 = {0, 0, 0, 0};
  i32x8 z8 = {0, 0, 0, 0, 0, 0, 0, 0};
  // 6-arg form (amdgpu-toolchain / clang-23): (g0, g1, g2, g3, g4, cpol)
  __builtin_amdgcn_tensor_load_to_lds(g0, g1, z4, z4, z8, 0);
}
#endif

// ---------------------------------------------------------------------------
// Kernel 2: fused broadcast-add + tanh + w-reduction.
// Block: 256 threads; tile = TS(=32) src rows x TTI(=16) tgt rows of one batch.
// Thread (s_local = tid>>3, t_local = tid&7) produces outputs for t_local and
// t_local+8 at s_local. LDS pitch 780 -> rows land on distinct banks
// (780 mod 64 = 12) for the 4 s-rows / 16 t-rows alive per wave.
// ---------------------------------------------------------------------------
__global__ __launch_bounds__(256) void fused_tanh_reduce(
    const float* __restrict__ src_lin,   // [B*S, H]
    const float* __restrict__ tgt_lin,   // [B*T, H]
    const float* __restrict__ w_res,     // [H]
    const float* __restrict__ b_res,     // [1]
    float* __restrict__ out)             // [B, T, S]
{
  __shared__ __align__(16) float sS[TS * RPITCH];
  __shared__ __align__(16) float sT[TTI * RPITCH];
  __shared__ __align__(16) float sW[HH];

  const int tid = threadIdx.x;
  const int b  = blockIdx.z;
  const int s0 = blockIdx.x * TS;
  const int t0 = blockIdx.y * TTI;
  const float* gsrc = src_lin + ((size_t)b * SS + s0) * HH;
  const float* gtgt = tgt_lin + ((size_t)b * TT_ + t0) * HH;

  for (int i = tid; i < HH; i += 256) sW[i] = w_res[i];

#if USE_TDM
  if (tid < 32) {  // wave 0 programs the DMA engine
    tdm_load_2d(gsrc, (u32)(size_t)&sS[0], TS,  (u32)(SS - s0));
    tdm_load_2d(gtgt, (u32)(size_t)&sT[0], TTI, (u32)(TT_ - t0));
    __builtin_amdgcn_s_wait_tensorcnt(0);
  }
  __syncthreads();
#else
  for (int idx = tid; idx < TS * (HH / 4); idx += 256) {
    int row = idx / (HH / 4);
    int dd  = (idx - row * (HH / 4)) * 4;
    float4 v = *(const float4*)(gsrc + (size_t)row * HH + dd);
    *(float4*)&sS[row * RPITCH + (dd >> 8) * SEGP + (dd & 255)] = v;
  }
  for (int idx = tid; idx < TTI * (HH / 4); idx += 256) {
    int row = idx / (HH / 4);
    int dd  = (idx - row * (HH / 4)) * 4;
    float4 v = *(const float4*)(gtgt + (size_t)row * HH + dd);
    *(float4*)&sT[row * RPITCH + (dd >> 8) * SEGP + (dd & 255)] = v;
  }
  __syncthreads();
#endif

  const int s_local = tid >> 3;
  const int t_local = tid & 7;
  float acc0 = 0.f, acc1 = 0.f;
  const float* sprow  = &sS[s_local * RPITCH];
  const float* tprow0 = &sT[t_local * RPITCH];
  const float* tprow1 = &sT[(t_local + 8) * RPITCH];

  for (int seg = 0; seg < 3; ++seg) {
    const float4* sp  = (const float4*)(sprow  + seg * SEGP);
    const float4* tp0 = (const float4*)(tprow0 + seg * SEGP);
    const float4* tp1 = (const float4*)(tprow1 + seg * SEGP);
    const float4* wp  = (const float4*)(&sW[seg * 256]);
#pragma unroll 4
    for (int i = 0; i < 64; ++i) {
      float4 sv = sp[i], a0 = tp0[i], a1 = tp1[i], wv = wp[i];
      acc0 += htanh(sv.x + a0.x) * wv.x;
      acc0 += htanh(sv.y + a0.y) * wv.y;
      acc0 += htanh(sv.z + a0.z) * wv.z;
      acc0 += htanh(sv.w + a0.w) * wv.w;
      acc1 += htanh(sv.x + a1.x) * wv.x;
      acc1 += htanh(sv.y + a1.y) * wv.y;
      acc1 += htanh(sv.z + a1.z) * wv.z;
      acc1 += htanh(sv.w + a1.w) * wv.w;
    }
  }

  const float bias = b_res[0];
  const int t = t0 + t_local;
  const int s = s0 + s_local;
  out[((size_t)b * TT_ + t) * SS + s]       = acc0 + bias;
  out[((size_t)b * TT_ + t + 8) * SS + s]   = acc1 + bias;
}

extern "C" void kernel_launch(void* const* d_in, const int* in_sizes, int n_in,
                              void* d_out, int out_size, void* d_ws, size_t ws_size,
                              hipStream_t stream) {
  const float* source = (const float*)d_in[0];  // [B,S,H]
  const float* target = (const float*)d_in[1];  // [B,T,H]
  const float* W_src  = (const float*)d_in[2];  // [H,H]
  const float* W_tgt  = (const float*)d_in[3];  // [H,H]
  const float* w_res  = (const float*)d_in[4];  // [1,H]
  const float* b_res  = (const float*)d_in[5];  // [1]
  float* out = (float*)d_out;                   // [B,T,S]

  float* src_lin = (float*)d_ws;                       // B*S*H f32
  float* tgt_lin = src_lin + (size_t)BB * SS * HH;     // B*T*H f32

  // src_lin = source @ W_src^T   (M = B*S = 1024)
  linear_wmma_f32<<<dim3((BB * SS) / 16, HH / 16), 32, 0, stream>>>(
      source, W_src, src_lin, BB * SS, HH);
  // tgt_lin = target @ W_tgt^T   (M = B*T = 256)
  linear_wmma_f32<<<dim3((BB * TT_) / 16, HH / 16), 32, 0, stream>>>(
      target, W_tgt, tgt_lin, BB * TT_, HH);

  // fused tanh + reduce: grid (S/TS, T/TTI, B)
  fused_tanh_reduce<<<dim3(SS / TS, TT_ / TTI, BB), 256, 0, stream>>>(
      src_lin, tgt_lin, w_res, b_res, out);
}